// Loss_87419764343124
// MI455X (gfx1250) — compile-verified
//
#include <hip/hip_runtime.h>
#include <stdint.h>

// GMM loss, B=131072, K=8, D=6.
//   d_in[0] means   [B,K,D]   f32      25 MB   (async -> LDS, NT)
//   d_in[1] covs    [B,K,D,D] f32     151 MB   (async -> LDS, NT)
//   d_in[2] weights [B,K]     f32       4 MB   (direct, fully coalesced)
//   d_in[3] target  [B,D]     f32       3 MB   (direct, cache-served x8)
//   d_in[4] label   [B]       i32     0.5 MB   (direct)
//   d_out: 4 f32 (loss, entropy_weight, entropy_dist, usage_entropy)
// Memory-bound: ~184 MB in / 16 B out -> ~8 us floor at 23.3 TB/s.
// VALU trimmed (hw log/exp/rsq/rcp, logdet via pivot product) to sit under that floor.

#define TPB 256
#define KC  8

static constexpr float EPSF               = 1e-6f;
static constexpr float HALF_D_LOG_2PI     = 5.513631199228037f;   // 0.5*6*log(2*pi)
static constexpr float D_ONE_PLUS_LOG_2PI = 17.027262398456074f;  // 6*(1+log(2*pi))

// Low 32 bits of a generic pointer to a __shared__ object == LDS byte offset.
__device__ __forceinline__ uint32_t lds_addr32(const void* p) {
  return (uint32_t)(uintptr_t)p;
}

// CDNA5 async copy: global -> LDS, 16B per lane, ASYNCcnt-tracked, non-temporal.
__device__ __forceinline__ void async_copy_b128_nt(uint32_t lds_off, const void* gaddr) {
  asm volatile("global_load_async_to_lds_b128 %0, %1, off th:TH_LOAD_NT"
               :: "v"(lds_off), "v"(gaddr)
               : "memory");
}

__device__ __forceinline__ void wait_async_all() {
  asm volatile("s_wait_asynccnt 0" ::: "memory");
}

__global__ __launch_bounds__(TPB) void gmm_loss_partial(
    const float* __restrict__ means,
    const float* __restrict__ covs,
    const float* __restrict__ weights,
    const float* __restrict__ target,
    const int*   __restrict__ label,
    float* __restrict__ ws,
    long long nPairs, int nBlocks)
{
  __shared__ __align__(16) float tileC[TPB * 36];  // 36 KB: 256 covariances
  __shared__ __align__(16) float tileM[TPB * 6];   //  6 KB: 256 means
  __shared__ float xw[8][12];                      // cross-wave partials

  const int tid  = threadIdx.x;
  const int lane = tid & 31;
  const int wv   = tid >> 5;
  const long long tileBase = (long long)blockIdx.x * TPB;
  const long long p   = tileBase + tid;      // pair index b*K + k
  const bool act      = p < nPairs;
  const bool fullTile = (tileBase + TPB) <= nPairs;

  // ---- issue async DMA for the big streams (full tiles only) ----
  if (fullTile) {
    const char*    gC = (const char*)covs + (size_t)tileBase * 144;  // 36 f32 each
    const uint32_t lC = lds_addr32(&tileC[0]);
#pragma unroll
    for (int i = 0; i < 9; ++i) {
      const int chunk = tid + i * TPB;                 // 2304 x 16B, contiguous
      async_copy_b128_nt(lC + (uint32_t)chunk * 16, gC + (size_t)chunk * 16);
    }
    const char*    gM = (const char*)means + (size_t)tileBase * 24;  // 6 f32 each
    const uint32_t lM = lds_addr32(&tileM[0]);
    async_copy_b128_nt(lM + (uint32_t)tid * 16, gM + (size_t)tid * 16);
    if (tid < TPB / 2) {
      const int chunk = TPB + tid;                     // 384 x 16B total
      async_copy_b128_nt(lM + (uint32_t)chunk * 16, gM + (size_t)chunk * 16);
    }
  }

  // ---- small direct loads overlap with the DMA ----
  float tgt[6];
  float w;
  int   lab;
  if (act) {
    const long long b = p >> 3;                 // K == 8
    const float2* tp = (const float2*)(target + (size_t)b * 6);
#pragma unroll
    for (int i = 0; i < 3; ++i) { const float2 t2 = tp[i]; tgt[2*i] = t2.x; tgt[2*i+1] = t2.y; }
    w   = weights[p];
    lab = label[b];
  } else {
#pragma unroll
    for (int i = 0; i < 6; ++i) tgt[i] = 0.0f;
    w = 0.125f; lab = 1;
  }

  // ---- gather this thread's covariance + mean ----
  float c[36];
  float diff[6];
  if (fullTile) {
    wait_async_all();
    __syncthreads();
    const float4* row = (const float4*)(tileC + tid * 36);  // 144B stride, 16B aligned
#pragma unroll
    for (int i = 0; i < 9; ++i) {
      const float4 v = row[i];
      c[4*i+0] = v.x; c[4*i+1] = v.y; c[4*i+2] = v.z; c[4*i+3] = v.w;
    }
    const float2* mrow = (const float2*)(tileM + tid * 6);  // 24B stride, 8B aligned
#pragma unroll
    for (int i = 0; i < 3; ++i) {
      const float2 m2 = mrow[i];
      diff[2*i+0] = tgt[2*i+0] - m2.x;
      diff[2*i+1] = tgt[2*i+1] - m2.y;
    }
  } else {
    if (act) {
      const float4* g = (const float4*)(covs + (size_t)p * 36);
#pragma unroll
      for (int i = 0; i < 9; ++i) {
        const float4 v = g[i];
        c[4*i+0] = v.x; c[4*i+1] = v.y; c[4*i+2] = v.z; c[4*i+3] = v.w;
      }
      const float2* mp = (const float2*)(means + (size_t)p * 6);
#pragma unroll
      for (int i = 0; i < 3; ++i) {
        const float2 m2 = mp[i];
        diff[2*i+0] = tgt[2*i+0] - m2.x;
        diff[2*i+1] = tgt[2*i+1] - m2.y;
      }
    } else {
#pragma unroll
      for (int i = 0; i < 36; ++i) c[i] = 0.0f;
      c[0] = c[7] = c[14] = c[21] = c[28] = c[35] = 1.0f;  // identity, benign math
#pragma unroll
      for (int i = 0; i < 6; ++i) diff[i] = 0.0f;
    }
  }

  // ---- in-register Cholesky (lower). Pivots s_j are bounded by the 0.5*I
  // regularization (lambda_min >= 0.5), so prod(s_j) stays in f32 range:
  // logdet = log(prod s_j) -- one hw log instead of six. 1/sqrt via v_rsq_f32.
  float invd[6];
  float pivprod = 1.0f;
#pragma unroll
  for (int j = 0; j < 6; ++j) {
    float s = c[j*6+j];
#pragma unroll
    for (int m = 0; m < j; ++m) s -= c[j*6+m] * c[j*6+m];
    s = fmaxf(s, 1e-30f);
    pivprod *= s;
    const float id = __builtin_amdgcn_rsqf(s);   // v_rsq_f32
    invd[j] = id;
#pragma unroll
    for (int i2 = j + 1; i2 < 6; ++i2) {
      float t = c[i2*6+j];
#pragma unroll
      for (int m = 0; m < j; ++m) t -= c[i2*6+m] * c[j*6+m];
      c[i2*6+j] = t * id;
    }
  }
  const float logdet = __logf(pivprod);

  // forward solve L y = diff; Mahalanobis = |y|^2
  float y[6];
  float maha = 0.0f;
#pragma unroll
  for (int j = 0; j < 6; ++j) {
    float t = diff[j];
#pragma unroll
    for (int m = 0; m < j; ++m) t -= c[j*6+m] * y[m];
    y[j] = t * invd[j];
    maha += y[j] * y[j];
  }

  const float lpc = -0.5f * (maha + logdet) - HALF_D_LOG_2PI + __logf(w + EPSF);

  // ---- logsumexp over the 8 components of this sample (k == lane%8) ----
  float mx = lpc;
  mx = fmaxf(mx, __shfl_xor(mx, 1, 8));
  mx = fmaxf(mx, __shfl_xor(mx, 2, 8));
  mx = fmaxf(mx, __shfl_xor(mx, 4, 8));
  const float e = __expf(lpc - mx);              // arg <= 0
  float se = e;
  se += __shfl_xor(se, 1, 8);
  se += __shfl_xor(se, 2, 8);
  se += __shfl_xor(se, 4, 8);
  const float log_prob = mx + __logf(se);        // se in [1, 8]
  const float resp     = e * __builtin_amdgcn_rcpf(se);  // v_rcp_f32

  const float labelf = (float)lab;              // NEG_WEIGHT == 1.0
  const float validf = (lab > 0) ? 1.0f : 0.0f;
  const bool  kzero  = ((tid & 7) == 0);        // once per sample b

  float a_loss  = (act && kzero) ? log_prob * labelf : 0.0f;
  float a_valid = (act && kzero) ? validf : 0.0f;
  float a_ew    = act ? w * __logf(w + EPSF) : 0.0f;
  float a_ed    = act ? w * 0.5f * (D_ONE_PLUS_LOG_2PI + logdet) : 0.0f;
  float a_us    = act ? resp * validf : 0.0f;

  // wave32 reductions (inactive lanes contribute 0; all lanes shuffle)
#pragma unroll
  for (int o = 1; o < 32; o <<= 1) {
    a_loss  += __shfl_xor(a_loss,  o, 32);
    a_valid += __shfl_xor(a_valid, o, 32);
    a_ew    += __shfl_xor(a_ew,    o, 32);
    a_ed    += __shfl_xor(a_ed,    o, 32);
  }
  a_us += __shfl_xor(a_us,  8, 32);   // sum lanes with equal lane%8
  a_us += __shfl_xor(a_us, 16, 32);

  if (lane < 8) xw[wv][lane] = a_us;  // lane == k
  if (lane == 0) { xw[wv][8] = a_loss; xw[wv][9] = a_ew; xw[wv][10] = a_ed; xw[wv][11] = a_valid; }
  __syncthreads();
  if (tid < 12) {
    float t = 0.0f;
#pragma unroll
    for (int i = 0; i < 8; ++i) t += xw[i][tid];
    // ws layout: [acc 0..11][block]  (0..7 usage[k], 8 loss, 9 ew, 10 ed, 11 valid)
    ws[(size_t)tid * nBlocks + blockIdx.x] = t;
  }
}

__global__ __launch_bounds__(384) void gmm_loss_final(
    const float* __restrict__ ws, float* __restrict__ out, int nBlocks, float invB)
{
  __shared__ float tot[12];
  const int wv   = threadIdx.x >> 5;   // 12 waves, one per accumulator
  const int lane = threadIdx.x & 31;
  float s = 0.0f;
  for (int i = lane; i < nBlocks; i += 32) s += ws[(size_t)wv * nBlocks + i];
#pragma unroll
  for (int o = 1; o < 32; o <<= 1) s += __shfl_xor(s, o, 32);
  if (lane == 0) tot[wv] = s;
  __syncthreads();
  if (threadIdx.x == 0) {
    const float sum_loss = tot[8];
    const float entropy_weight = -tot[9]  * invB;
    const float entropy_dist   = -tot[10] * invB;
    const float num_valid = tot[11] + EPSF;
    float usage_entropy = 0.0f;
#pragma unroll
    for (int k = 0; k < KC; ++k) {
      const float au = tot[k] / num_valid;
      usage_entropy -= au * logf(au + EPSF);
    }
    const float loss = -(sum_loss * invB)
                       - 0.1f  * entropy_weight
                       - 0.01f * entropy_dist
                       - 0.01f * usage_entropy;
    out[0] = loss;
    out[1] = entropy_weight;
    out[2] = entropy_dist;
    out[3] = usage_entropy;
  }
}

extern "C" void kernel_launch(void* const* d_in, const int* in_sizes, int n_in,
                              void* d_out, int out_size, void* d_ws, size_t ws_size,
                              hipStream_t stream) {
  const float* means   = (const float*)d_in[0];
  const float* covs    = (const float*)d_in[1];
  const float* weights = (const float*)d_in[2];
  const float* target  = (const float*)d_in[3];
  const int*   label   = (const int*)d_in[4];
  const int B = in_sizes[4];                       // label count
  const long long nPairs = (long long)B * KC;
  const int nBlocks = (int)((nPairs + TPB - 1) / TPB);
  float* ws = (float*)d_ws;                        // needs 12*nBlocks floats (~196 KB)

  gmm_loss_partial<<<nBlocks, TPB, 0, stream>>>(means, covs, weights, target, label,
                                                ws, nPairs, nBlocks);
  gmm_loss_final<<<1, 384, 0, stream>>>(ws, (float*)d_out, nBlocks, 1.0f / (float)B);
}